// PonderRelationalGraphConvModel_66408784331252
// MI455X (gfx1250) — compile-verified
//
#include <hip/hip_runtime.h>
#include <hip/hip_bf16.h>

#define NN 30000
#define EE 600000
#define RR 8
#define BB 4
#define HH 128
#define OO 64
#define NPO 30064   // N + O

typedef __attribute__((ext_vector_type(2))) float v2f;
typedef __attribute__((ext_vector_type(8))) float v8f;

#define AP0 (OO + 4)   // padded LDS row stride, gemm0 A tile (68)
#define AP1 (HH + 4)   // padded LDS row stride, gemm1 A tile (132)

// ---------------------------------------------------------------------------
// Small weight prep: Wf0[r,o,f] = sum_b w_rel0[r,b]*w_bases0[b,N+o,f]
//                    W1 [r,i,o] = sum_b w_rel1[r,b]*w_bases1[b,i,o]
// ---------------------------------------------------------------------------
__global__ void make_weights(const float* __restrict__ w_rel0,
                             const float* __restrict__ w_bases0,
                             const float* __restrict__ w_rel1,
                             const float* __restrict__ w_bases1,
                             float* __restrict__ Wf0,
                             float* __restrict__ W1) {
  int i = blockIdx.x * blockDim.x + threadIdx.x;   // 0 .. 65535
  if (i < RR * OO * HH) {
    int r = i / (OO * HH);
    int rem = i % (OO * HH);
    int o = rem / HH;
    int f = rem % HH;
    float s = 0.f;
    #pragma unroll
    for (int b = 0; b < BB; ++b)
      s += w_rel0[r * BB + b] *
           w_bases0[(size_t)b * NPO * HH + (size_t)(NN + o) * HH + f];
    Wf0[i] = s;
  }
  if (i < RR * HH * OO) {
    int r = i / (HH * OO);
    int rem = i % (HH * OO);
    int ii = rem / OO;
    int o = rem % OO;
    float s = 0.f;
    #pragma unroll
    for (int b = 0; b < BB; ++b)
      s += w_rel1[r * BB + b] * w_bases1[(size_t)b * HH * OO + ii * OO + o];
    W1[i] = s;
  }
}

// ---------------------------------------------------------------------------
// Layer-0 GEMM with WMMA f32 16x16x4:
//   G[n, c*16+idx] = sum_o h[n,o]*Wf0[r,o,f] + sum_b w_rel0[r,b]*w_bases0[b,n,f]
// c = r*8 + f0/16 indexes 64 column tiles of the 1024-wide G row.
// Grid: (N/16, 64/4), block 128 (4 waves, one 16x16 tile each).
// A tile (16x64) staged block-wide; B slab (64 rows x 16 cols) staged
// per-wave *transposed* in LDS -> zero VMEM in the WMMA loop.
// ---------------------------------------------------------------------------
template <int USE_ONES>
__global__ void gemm0_wmma(const float* __restrict__ h,      // (N,64)
                           const float* __restrict__ Wf0,    // (8,64,128)
                           const float* __restrict__ w_rel0, // (8,4)
                           const float* __restrict__ w_bases0,
                           float* __restrict__ G) {          // (N, 1024)
  const int lane = threadIdx.x & 31;
  const int wave = threadIdx.x >> 5;
  const int n0   = blockIdx.x * 16;
  const int c    = blockIdx.y * 4 + wave;   // 0..63
  const int r    = c >> 3;
  const int f0   = (c & 7) * 16;
  const int half = lane >> 4;               // 0 or 1
  const int idx  = lane & 15;

  __shared__ float As[16 * AP0];            // block-shared A tile
  __shared__ float Bs[4][16 * AP0];         // per-wave transposed B slab

  if (!USE_ONES) {
    // cooperative stage: 128 threads x 8 floats = 16x64 tile
    const int t    = threadIdx.x;
    const int row  = t >> 3;
    const int col0 = (t & 7) * 8;
    const float* hp = h + (size_t)(n0 + row) * OO + col0;
    float4 p0 = *(const float4*)(hp);
    float4 p1 = *(const float4*)(hp + 4);
    *(float4*)&As[row * AP0 + col0]     = p0;
    *(float4*)&As[row * AP0 + col0 + 4] = p1;
  }

  // stage B transposed: Bs[w][f_local*AP0 + k] = Wf0[r, k, f0+f_local]
  {
    const float* Br = Wf0 + (size_t)r * OO * HH;
    const int krow = lane >> 2;        // 0..7
    const int fcol = (lane & 3) * 4;   // 0,4,8,12
    #pragma unroll
    for (int k0 = 0; k0 < OO; k0 += 8) {
      const int k = k0 + krow;
      float4 p = *(const float4*)(Br + (size_t)k * HH + f0 + fcol);
      Bs[wave][(fcol + 0) * AP0 + k] = p.x;
      Bs[wave][(fcol + 1) * AP0 + k] = p.y;
      Bs[wave][(fcol + 2) * AP0 + k] = p.z;
      Bs[wave][(fcol + 3) * AP0 + k] = p.w;
    }
  }
  __syncthreads();

  v8f acc = {};
  #pragma unroll
  for (int k0 = 0; k0 < OO; k0 += 4) {
    const int ka = k0 + 2 * half;
    v2f a;
    if (USE_ONES) {
      a.x = 1.0f; a.y = 1.0f;
    } else {
      a = *(const v2f*)&As[idx * AP0 + ka];
    }
    v2f b = *(const v2f*)&Bs[wave][idx * AP0 + ka];
    acc = __builtin_amdgcn_wmma_f32_16x16x4_f32(false, a, false, b,
                                                (short)0, acc, false, false);
  }

  float wr[BB];
  #pragma unroll
  for (int b = 0; b < BB; ++b) wr[b] = w_rel0[r * BB + b];

  #pragma unroll
  for (int j = 0; j < 8; ++j) {
    const int m = j + 8 * half;
    const int n = n0 + m;
    const int f = f0 + idx;
    float wid = 0.f;
    #pragma unroll
    for (int b = 0; b < BB; ++b)
      wid += wr[b] * w_bases0[(size_t)b * NPO * HH + (size_t)n * HH + f];
    G[(size_t)n * 1024 + c * 16 + idx] = acc[j] + wid;
  }
}

// ---------------------------------------------------------------------------
// Layer-1 GEMM with WMMA f32 16x16x4 (ReLU fused into LDS staging):
//   hr1[n, c*16+idx] = sum_i relu(acc0[n,i]) * W1[r,i,o]
// c = r*4 + o0/16 indexes 32 column tiles of the 512-wide hr1 row.
// Grid: (N/16, 32/4), block 128. A tile + transposed B slab in LDS.
// ---------------------------------------------------------------------------
__global__ void gemm1_wmma(const float* __restrict__ acc0,  // (N,128)
                           const float* __restrict__ W1,    // (8,128,64)
                           float* __restrict__ hr1) {       // (N, 512)
  const int lane = threadIdx.x & 31;
  const int wave = threadIdx.x >> 5;
  const int n0   = blockIdx.x * 16;
  const int c    = blockIdx.y * 4 + wave;   // 0..31
  const int r    = c >> 2;
  const int o0   = (c & 3) * 16;
  const int half = lane >> 4;
  const int idx  = lane & 15;

  __shared__ float As[16 * AP1];            // block-shared A tile (ReLU'd)
  __shared__ float Bs[4][16 * AP1];         // per-wave transposed B slab

  {
    // cooperative stage with fused ReLU: 128 threads x 16 floats = 16x128
    const int t    = threadIdx.x;
    const int row  = t >> 3;
    const int col0 = (t & 7) * 16;
    const float* ap = acc0 + (size_t)(n0 + row) * HH + col0;
    float* sp = &As[row * AP1 + col0];
    #pragma unroll
    for (int q = 0; q < 4; ++q) {
      float4 p = *(const float4*)(ap + q * 4);
      p.x = fmaxf(p.x, 0.f); p.y = fmaxf(p.y, 0.f);
      p.z = fmaxf(p.z, 0.f); p.w = fmaxf(p.w, 0.f);
      *(float4*)(sp + q * 4) = p;
    }
  }

  // stage B transposed: Bs[w][o_local*AP1 + k] = W1[r, k, o0+o_local]
  {
    const float* Br = W1 + (size_t)r * HH * OO;
    const int krow = lane >> 2;        // 0..7
    const int ocol = (lane & 3) * 4;   // 0,4,8,12
    #pragma unroll
    for (int k0 = 0; k0 < HH; k0 += 8) {
      const int k = k0 + krow;
      float4 p = *(const float4*)(Br + (size_t)k * OO + o0 + ocol);
      Bs[wave][(ocol + 0) * AP1 + k] = p.x;
      Bs[wave][(ocol + 1) * AP1 + k] = p.y;
      Bs[wave][(ocol + 2) * AP1 + k] = p.z;
      Bs[wave][(ocol + 3) * AP1 + k] = p.w;
    }
  }
  __syncthreads();

  v8f acc = {};
  #pragma unroll
  for (int k0 = 0; k0 < HH; k0 += 4) {
    const int ka = k0 + 2 * half;
    v2f a = *(const v2f*)&As[idx * AP1 + ka];
    v2f b = *(const v2f*)&Bs[wave][idx * AP1 + ka];
    acc = __builtin_amdgcn_wmma_f32_16x16x4_f32(false, a, false, b,
                                                (short)0, acc, false, false);
  }

  #pragma unroll
  for (int j = 0; j < 8; ++j) {
    const int m = j + 8 * half;
    hr1[(size_t)(n0 + m) * 512 + c * 16 + idx] = acc[j];
  }
}

// ---------------------------------------------------------------------------
// Edge scatter, layer 0: acc0[dst, 0:128] += val * G[src, rel*128 : rel*128+128]
// One wave per edge; edge index forced wave-uniform (scalar metadata loads);
// float4 gather (512B/wave) + f32 global atomics.
// ---------------------------------------------------------------------------
__global__ void edge_scatter0(const int* __restrict__ src,
                              const int* __restrict__ dst,
                              const int* __restrict__ rel,
                              const float* __restrict__ val,
                              const float* __restrict__ G,
                              float* __restrict__ acc0) {
  const int gid  = blockIdx.x * blockDim.x + threadIdx.x;
  const int e    = __builtin_amdgcn_readfirstlane(gid >> 5);
  const int lane = gid & 31;
  if (e >= EE) return;
  __builtin_prefetch(&src[e + 256], 0, 1);
  __builtin_prefetch(&dst[e + 256], 0, 1);
  const int   s = src[e];
  const int   d = dst[e];
  const int   r = rel[e];
  const float v = val[e];
  const float4* g4 = (const float4*)(G + (size_t)s * 1024 + r * HH);
  float4 gv = g4[lane];
  float* op = acc0 + (size_t)d * HH + lane * 4;
  atomicAdd(op + 0, v * gv.x);
  atomicAdd(op + 1, v * gv.y);
  atomicAdd(op + 2, v * gv.z);
  atomicAdd(op + 3, v * gv.w);
}

// ---------------------------------------------------------------------------
// Edge scatter, layer 1: hout[dst, 0:64] += val * hr1[src, rel*64 : rel*64+64]
// ---------------------------------------------------------------------------
__global__ void edge_scatter1(const int* __restrict__ src,
                              const int* __restrict__ dst,
                              const int* __restrict__ rel,
                              const float* __restrict__ val,
                              const float* __restrict__ hr1,
                              float* __restrict__ hout) {
  const int gid  = blockIdx.x * blockDim.x + threadIdx.x;
  const int e    = __builtin_amdgcn_readfirstlane(gid >> 5);
  const int lane = gid & 31;
  if (e >= EE) return;
  __builtin_prefetch(&src[e + 256], 0, 1);
  __builtin_prefetch(&dst[e + 256], 0, 1);
  const int   s = src[e];
  const int   d = dst[e];
  const int   r = rel[e];
  const float v = val[e];
  const float2* g2 = (const float2*)(hr1 + (size_t)s * 512 + r * OO);
  float2 gv = g2[lane];
  float* op = hout + (size_t)d * OO + lane * 2;
  atomicAdd(op + 0, v * gv.x);
  atomicAdd(op + 1, v * gv.y);
}

// ---------------------------------------------------------------------------
// Ponder head: lam = sigmoid(h_a @ w_lambda + b); write ps/lams for both steps.
// ---------------------------------------------------------------------------
__global__ void finalize(const float* __restrict__ h_a,
                         const float* __restrict__ w_lambda,
                         const float* __restrict__ b_lambda,
                         float* __restrict__ ps,
                         float* __restrict__ lams) {
  const int n = blockIdx.x * blockDim.x + threadIdx.x;
  if (n >= NN) return;
  float acc = b_lambda[0];
  const float4* hv = (const float4*)(h_a + (size_t)n * OO);
  const float4* wv = (const float4*)w_lambda;
  #pragma unroll
  for (int q = 0; q < OO / 4; ++q) {
    float4 hq = hv[q];
    float4 wq = wv[q];
    acc += hq.x * wq.x + hq.y * wq.y + hq.z * wq.z + hq.w * wq.w;
  }
  const float lam = 1.0f / (1.0f + __expf(-acc));
  ps[n]        = lam;
  lams[n]      = lam;
  ps[NN + n]   = 1.0f - lam;
  lams[NN + n] = 1.0f;
}

// ---------------------------------------------------------------------------
extern "C" void kernel_launch(void* const* d_in, const int* in_sizes, int n_in,
                              void* d_out, int out_size, void* d_ws, size_t ws_size,
                              hipStream_t stream) {
  const int*   edge_src  = (const int*)d_in[0];
  const int*   edge_dst  = (const int*)d_in[1];
  const int*   edge_rel  = (const int*)d_in[2];
  const float* edge_val  = (const float*)d_in[3];
  const float* w_bases0  = (const float*)d_in[4];
  const float* w_rel0    = (const float*)d_in[5];
  const float* w_bases1  = (const float*)d_in[6];
  const float* w_rel1    = (const float*)d_in[7];
  const float* w_lambda  = (const float*)d_in[8];
  const float* b_lambda  = (const float*)d_in[9];

  float* out = (float*)d_out;
  float* ys0  = out;                       // (N,64) = h_a
  float* ys1  = out + (size_t)NN * OO;     // (N,64) = h_b
  float* ps   = out + (size_t)2 * NN * OO; // (2,N)
  float* lams = ps + (size_t)2 * NN;       // (2,N)

  float* ws   = (float*)d_ws;
  float* G    = ws;                            // (N,1024); hr1 (N,512) aliases it
  float* acc0 = G + (size_t)NN * 1024;         // (N,128)
  float* Wf0  = acc0 + (size_t)NN * HH;        // 8*64*128
  float* W1   = Wf0 + RR * OO * HH;            // 8*128*64
  float* hr1  = G;                             // alias: G dead when hr1 live

  dim3 blk(128);
  dim3 g0(NN / 16, 64 / 4);   // gemm0 tiles
  dim3 g1(NN / 16, 32 / 4);   // gemm1 tiles
  const int edgeBlocks = (EE * 32) / 256;

  make_weights<<<(RR * HH * OO + 255) / 256, 256, 0, stream>>>(
      w_rel0, w_bases0, w_rel1, w_bases1, Wf0, W1);

  // ---- gnn #1: h_a = gnn(ones) ----
  gemm0_wmma<1><<<g0, blk, 0, stream>>>(nullptr, Wf0, w_rel0, w_bases0, G);
  hipMemsetAsync(acc0, 0, (size_t)NN * HH * sizeof(float), stream);
  edge_scatter0<<<edgeBlocks, 256, 0, stream>>>(edge_src, edge_dst, edge_rel,
                                                edge_val, G, acc0);
  gemm1_wmma<<<g1, blk, 0, stream>>>(acc0, W1, hr1);
  hipMemsetAsync(ys0, 0, (size_t)NN * OO * sizeof(float), stream);
  edge_scatter1<<<edgeBlocks, 256, 0, stream>>>(edge_src, edge_dst, edge_rel,
                                                edge_val, hr1, ys0);

  // ---- gnn #2: h_b = gnn(h_a) ----
  gemm0_wmma<0><<<g0, blk, 0, stream>>>(ys0, Wf0, w_rel0, w_bases0, G);
  hipMemsetAsync(acc0, 0, (size_t)NN * HH * sizeof(float), stream);
  edge_scatter0<<<edgeBlocks, 256, 0, stream>>>(edge_src, edge_dst, edge_rel,
                                                edge_val, G, acc0);
  gemm1_wmma<<<g1, blk, 0, stream>>>(acc0, W1, hr1);
  hipMemsetAsync(ys1, 0, (size_t)NN * OO * sizeof(float), stream);
  edge_scatter1<<<edgeBlocks, 256, 0, stream>>>(edge_src, edge_dst, edge_rel,
                                                edge_val, hr1, ys1);

  // ---- ponder head ----
  finalize<<<(NN + 255) / 256, 256, 0, stream>>>(ys0, w_lambda, b_lambda, ps, lams);
}